// DomainCalibratedLoss_89704686944429
// MI455X (gfx1250) — compile-verified
//
#include <hip/hip_runtime.h>
#include <hip/hip_bf16.h>

typedef float v2f __attribute__((ext_vector_type(2)));
typedef float v8f __attribute__((ext_vector_type(8)));

#define NPTS   500000
#define NCLS   200
#define NDOM   8
#define TILE_M 16
#define NTILES (NPTS / TILE_M)   // 31250 (exact)
#define KSTEPS (NCLS / 4)        // 50 WMMA steps of K=4
#define BLK    256
#define WPB    (BLK / 32)        // waves per block (wave32)
#define NBLOCKS 1024

// Phase 1: per-block partial loss sums. One wave computes one 16-point tile via
// V_WMMA_F32_16X16X4_F32:  C[16x16] += expA[16x4] x Wt[4x16]  (8 real domain cols).
__global__ __launch_bounds__(BLK) void dcl_partial_kernel(
    const float* __restrict__ inputs,   // [N, C] logits
    const int*   __restrict__ targets,  // [N]
    const int*   __restrict__ domains,  // [N]
    const float* __restrict__ w,        // [D, C] positive counts
    float*       __restrict__ partials) // [gridDim.x]
{
    __shared__ float lds_w[NCLS * 16];  // W transposed: [c][n], cols 8..15 zero-padded
    __shared__ float lds_red[BLK];

    const int tid = threadIdx.x;

    // Stage W^T into LDS with zero padding to 16 columns.
    for (int i = tid; i < NCLS * 16; i += BLK) lds_w[i] = 0.0f;
    __syncthreads();
    for (int i = tid; i < NDOM * NCLS; i += BLK) {
        const int d = i / NCLS;
        const int c = i - d * NCLS;
        lds_w[c * 16 + d] = w[i];
    }
    __syncthreads();

    const int lane = tid & 31;
    const int wave = tid >> 5;
    const int m    = lane & 15;         // A-row M / B-col N owned by this lane
    const int kh   = (lane >> 4) * 2;   // K sub-offset within each K=4 chunk
    const int tstride = gridDim.x * WPB;

    float acc_loss = 0.0f;

    for (int tile = blockIdx.x * WPB + wave; tile < NTILES; tile += tstride)
    {
        const long   base = (long)tile * TILE_M;
        const float* rowp = inputs + (base + m) * (long)NCLS;

        // Prefetch this lane's row of the NEXT tile (speculative, safe if OOB).
        // Row = 800B -> 7 x 128B lines.
        {
            const float* nrow = rowp + (long)tstride * TILE_M * NCLS;
            #pragma unroll
            for (int j = 0; j < 7; ++j)
                __builtin_prefetch(nrow + j * 32, 0, 0);
        }

        // Hoist per-point metadata loads so they overlap the WMMA loop.
        int t = 0, d = 0;
        float p_t = 0.0f;
        if (lane < 16) {
            t = targets[base + m];
            d = domains[base + m] & 7;
            int tcl = t;
            if (tcl < 0) tcl = 0;
            if (tcl > NCLS - 1) tcl = NCLS - 1;
            p_t = rowp[tcl];            // inputs[(base+m)*NCLS + clip(t)]
        }
        int tc = t;
        if (tc < 0) tc = 0;
        if (tc > NCLS - 1) tc = NCLS - 1;

        v8f acc = {0.f, 0.f, 0.f, 0.f, 0.f, 0.f, 0.f, 0.f};
        #pragma unroll 10
        for (int s = 0; s < KSTEPS; ++s) {
            const int k0 = s * 4;
            // A fragment: exp(logit) pair for this lane (8B aligned: 200 and k0+kh even)
            const v2f raw = *(const v2f*)(rowp + k0 + kh);
            v2f aval;
            aval.x = __expf(raw.x);
            aval.y = __expf(raw.y);
            // B fragment: W^T rows k0+kh, k0+kh+1 at column m (zero for m>=8 via padding)
            v2f bval;
            bval.x = lds_w[(k0 + kh)     * 16 + m];
            bval.y = lds_w[(k0 + kh + 1) * 16 + m];
            // (neg_a, A, neg_b, B, c_mod, C, reuse_a, reuse_b)
            acc = __builtin_amdgcn_wmma_f32_16x16x4_f32(
                false, aval, false, bval, (short)0, acc, false, false);
        }

        // Lane m (m<16) owns point m: gather S = C[m][d_m].
        // C[M][N] lives in lane N + 16*(M>=8), VGPR (M&7).
        const int src = d + ((m & 8) ? 16 : 0);
        float S = 0.0f;
        #pragma unroll
        for (int v = 0; v < 8; ++v) {
            const float g = __shfl(acc[v], src, 32);  // all lanes participate
            if ((m & 7) == v) S = g;
        }

        if (lane < 16) {
            const bool valid = (t != 255);
            const float lw_t = __logf(lds_w[tc * 16 + d]);  // log w[d, clip(t)] from LDS
            const float lse  = __logf(S);
            if (valid) acc_loss += (lse - lw_t - p_t);
        }
    }

    // Deterministic block reduction.
    lds_red[tid] = acc_loss;
    __syncthreads();
    for (int off = BLK / 2; off > 0; off >>= 1) {
        if (tid < off) lds_red[tid] += lds_red[tid + off];
        __syncthreads();
    }
    if (tid == 0) partials[blockIdx.x] = lds_red[0];
}

// Phase 2: fixed-order reduction of block partials -> mean loss.
__global__ __launch_bounds__(BLK) void dcl_finalize_kernel(
    const float* __restrict__ partials, float* __restrict__ out, int nparts)
{
    __shared__ float red[BLK];
    float s = 0.0f;
    for (int i = threadIdx.x; i < nparts; i += BLK) s += partials[i];
    red[threadIdx.x] = s;
    __syncthreads();
    for (int off = BLK / 2; off > 0; off >>= 1) {
        if (threadIdx.x < off) red[threadIdx.x] += red[threadIdx.x + off];
        __syncthreads();
    }
    if (threadIdx.x == 0) out[0] = red[0] / (float)NPTS;
}

extern "C" void kernel_launch(void* const* d_in, const int* in_sizes, int n_in,
                              void* d_out, int out_size, void* d_ws, size_t ws_size,
                              hipStream_t stream)
{
    const float* inputs  = (const float*)d_in[0];
    const int*   targets = (const int*)d_in[1];
    const int*   domains = (const int*)d_in[2];
    const float* w       = (const float*)d_in[3];
    float* partials = (float*)d_ws;   // NBLOCKS floats of scratch

    dcl_partial_kernel<<<NBLOCKS, BLK, 0, stream>>>(inputs, targets, domains, w, partials);
    dcl_finalize_kernel<<<1, BLK, 0, stream>>>(partials, (float*)d_out, NBLOCKS);
}